// ForgetMult_66194035966608
// MI455X (gfx1250) — compile-verified
//
#include <hip/hip_runtime.h>

namespace {

constexpr int SEQ = 2048;
constexpr int BH  = 16384;            // B*H = 16*1024
constexpr int CCH = 16;               // chunks over SEQ
constexpr int TCH = SEQ / CCH;        // 128 timesteps per chunk
constexpr int TT  = 4;                // timesteps per prefetch tile
constexpr int NTILES = TCH / TT;      // 32 tiles per chunk
constexpr int NBUF = 3;               // triple buffer (wave-private)
constexpr size_t WS_BYTES = 3ull * CCH * BH * sizeof(float);

typedef int v4i __attribute__((ext_vector_type(4)));
typedef __attribute__((address_space(1))) v4i glb_v4i;
typedef __attribute__((address_space(3))) v4i lds_v4i;

// Async global->LDS copy of 16 bytes per lane (ASYNCcnt-tracked).
__device__ __forceinline__ void async_cp16(const float* gp, float* lp) {
    __builtin_amdgcn_global_load_async_to_lds_b128(
        (glb_v4i*)gp, (lds_v4i*)lp, 0, 0);
}

__device__ __forceinline__ void wait_async16() { asm volatile("s_wait_asynccnt 16" ::: "memory"); }
__device__ __forceinline__ void wait_async8()  { asm volatile("s_wait_asynccnt 8"  ::: "memory"); }
__device__ __forceinline__ void wait_async0()  { asm volatile("s_wait_asynccnt 0"  ::: "memory"); }

// h = f*x + (1-f)*h, per component
__device__ __forceinline__ void rec_step(const float4& fv, const float4& xv, float4& h) {
    h.x = fmaf(fv.x, xv.x, (1.0f - fv.x) * h.x);
    h.y = fmaf(fv.y, xv.y, (1.0f - fv.y) * h.y);
    h.z = fmaf(fv.z, xv.z, (1.0f - fv.z) * h.z);
    h.w = fmaf(fv.w, xv.w, (1.0f - fv.w) * h.w);
}

// ---------------- Pass 1: per-chunk decay product + local end value ----------------
__global__ __launch_bounds__(256) void fm_pass1(const float* __restrict__ f,
                                                const float* __restrict__ x,
                                                float* __restrict__ Aout,
                                                float* __restrict__ hEnd) {
    // wave-private tiles: [wave][buf][tt][f/x][128 floats]
    __shared__ float smem[8][NBUF][TT][2][128];
    const int tid = threadIdx.x;
    const int w = tid >> 5, l = tid & 31;
    const int g = blockIdx.x;           // channel group 0..15 (1024 ch each)
    const int c = blockIdx.y;           // chunk 0..CCH-1
    const int chOff = g * 1024 + w * 128 + l * 4;
    const int t0 = c * TCH;
    const float* fb = f + (size_t)t0 * BH + chOff;
    const float* xb = x + (size_t)t0 * BH + chOff;

    auto issue = [&](int tile) {
        const int b = tile % NBUF;
        const size_t base = (size_t)tile * TT * BH;
#pragma unroll
        for (int tt = 0; tt < TT; ++tt) {
            async_cp16(fb + base + (size_t)tt * BH, &smem[w][b][tt][0][l * 4]);
            async_cp16(xb + base + (size_t)tt * BH, &smem[w][b][tt][1][l * 4]);
        }
    };

    issue(0);
    issue(1);

    float4 h    = make_float4(0.f, 0.f, 0.f, 0.f);
    float4 Aacc = make_float4(1.f, 1.f, 1.f, 1.f);

    for (int tile = 0; tile < NTILES; ++tile) {
        if (tile + 2 < NTILES) issue(tile + 2);
        if (tile + 2 < NTILES)      wait_async16();
        else if (tile + 1 < NTILES) wait_async8();
        else                        wait_async0();
        const int b = tile % NBUF;
#pragma unroll
        for (int tt = 0; tt < TT; ++tt) {
            float4 fv = *(const float4*)&smem[w][b][tt][0][l * 4];
            float4 xv = *(const float4*)&smem[w][b][tt][1][l * 4];
            float ax = 1.0f - fv.x, ay = 1.0f - fv.y, az = 1.0f - fv.z, aw = 1.0f - fv.w;
            h.x = fmaf(fv.x, xv.x, ax * h.x);
            h.y = fmaf(fv.y, xv.y, ay * h.y);
            h.z = fmaf(fv.z, xv.z, az * h.z);
            h.w = fmaf(fv.w, xv.w, aw * h.w);
            Aacc.x *= ax; Aacc.y *= ay; Aacc.z *= az; Aacc.w *= aw;
        }
    }
    *(float4*)(Aout + (size_t)c * BH + chOff) = Aacc;
    *(float4*)(hEnd + (size_t)c * BH + chOff) = h;
}

// ---------------- Pass 2: sequential combine over chunks (tiny) ----------------
__global__ __launch_bounds__(256) void fm_pass2(const float* __restrict__ A,
                                                const float* __restrict__ hEnd,
                                                const float* __restrict__ h0,
                                                float* __restrict__ carryIn) {
    const int q = blockIdx.x * blockDim.x + threadIdx.x;   // quad 0..4095
    const int off = q * 4;
    float4 carry = *(const float4*)(h0 + off);
    for (int c = 0; c < CCH; ++c) {
        *(float4*)(carryIn + (size_t)c * BH + off) = carry;
        float4 a = *(const float4*)(A    + (size_t)c * BH + off);
        float4 e = *(const float4*)(hEnd + (size_t)c * BH + off);
        carry.x = fmaf(a.x, carry.x, e.x);
        carry.y = fmaf(a.y, carry.y, e.y);
        carry.z = fmaf(a.z, carry.z, e.z);
        carry.w = fmaf(a.w, carry.w, e.w);
    }
}

// ---------------- Pass 3: re-scan each chunk with correct carry-in ----------------
__global__ __launch_bounds__(256) void fm_pass3(const float* __restrict__ f,
                                                const float* __restrict__ x,
                                                const float* __restrict__ carryIn,
                                                float* __restrict__ out) {
    __shared__ float smem[8][NBUF][TT][2][128];
    const int tid = threadIdx.x;
    const int w = tid >> 5, l = tid & 31;
    const int g = blockIdx.x;
    const int c = blockIdx.y;
    const int chOff = g * 1024 + w * 128 + l * 4;
    const int t0 = c * TCH;
    const float* fb = f + (size_t)t0 * BH + chOff;
    const float* xb = x + (size_t)t0 * BH + chOff;
    float* ob = out + (size_t)t0 * BH + chOff;

    auto issue = [&](int tile) {
        const int b = tile % NBUF;
        const size_t base = (size_t)tile * TT * BH;
#pragma unroll
        for (int tt = 0; tt < TT; ++tt) {
            async_cp16(fb + base + (size_t)tt * BH, &smem[w][b][tt][0][l * 4]);
            async_cp16(xb + base + (size_t)tt * BH, &smem[w][b][tt][1][l * 4]);
        }
    };

    issue(0);
    issue(1);

    float4 h = *(const float4*)(carryIn + (size_t)c * BH + chOff);

    for (int tile = 0; tile < NTILES; ++tile) {
        if (tile + 2 < NTILES) issue(tile + 2);
        if (tile + 2 < NTILES)      wait_async16();
        else if (tile + 1 < NTILES) wait_async8();
        else                        wait_async0();
        const int b = tile % NBUF;
        const size_t base = (size_t)tile * TT * BH;
#pragma unroll
        for (int tt = 0; tt < TT; ++tt) {
            float4 fv = *(const float4*)&smem[w][b][tt][0][l * 4];
            float4 xv = *(const float4*)&smem[w][b][tt][1][l * 4];
            rec_step(fv, xv, h);
            *(float4*)(ob + base + (size_t)tt * BH) = h;
        }
    }
}

// ---------------- Fallback: naive single-pass (if workspace too small) ----------------
__global__ __launch_bounds__(256) void fm_naive(const float* __restrict__ f,
                                                const float* __restrict__ x,
                                                const float* __restrict__ h0,
                                                float* __restrict__ out) {
    const int q = blockIdx.x * blockDim.x + threadIdx.x;   // quad 0..4095
    const int off = q * 4;
    float4 h = *(const float4*)(h0 + off);
    for (int t = 0; t < SEQ; ++t) {
        float4 fv = *(const float4*)(f + (size_t)t * BH + off);
        float4 xv = *(const float4*)(x + (size_t)t * BH + off);
        rec_step(fv, xv, h);
        *(float4*)(out + (size_t)t * BH + off) = h;
    }
}

} // anonymous namespace

extern "C" void kernel_launch(void* const* d_in, const int* in_sizes, int n_in,
                              void* d_out, int out_size, void* d_ws, size_t ws_size,
                              hipStream_t stream) {
    (void)in_sizes; (void)n_in; (void)out_size;
    const float* f  = (const float*)d_in[0];
    const float* x  = (const float*)d_in[1];
    const float* h0 = (const float*)d_in[2];
    float* out = (float*)d_out;

    if (ws_size >= WS_BYTES) {
        float* A       = (float*)d_ws;
        float* hEnd    = A + (size_t)CCH * BH;
        float* carryIn = hEnd + (size_t)CCH * BH;
        dim3 grid(BH / 1024, CCH);   // 16 channel groups x 16 chunks
        fm_pass1<<<grid, 256, 0, stream>>>(f, x, A, hEnd);
        fm_pass2<<<(BH / 4) / 256, 256, 0, stream>>>(A, hEnd, h0, carryIn);
        fm_pass3<<<grid, 256, 0, stream>>>(f, x, carryIn, out);
    } else {
        fm_naive<<<(BH / 4) / 256, 256, 0, stream>>>(f, x, h0, out);
    }
}